// EmbeddingLayer_79302276153442
// MI455X (gfx1250) — compile-verified
//
#include <hip/hip_runtime.h>
#include <hip/hip_bf16.h>
#include <stdint.h>

#define BATCH    1024
#define NUM_CONT 1028
#define EMB      128

typedef float v4f __attribute__((ext_vector_type(4)));

// ---------------------------------------------------------------------------
// Kernel 1: per-feature LayerNorm moments.
// For each feature f:  mW = mean(W[f,:]), mB = mean(b[f,:]),
//                      vW = Var(W), c2 = 2*Cov(W,b), vB = Var(b)
// One wave32 per feature; two-pass (centered) second moments for accuracy.
// Output: st[f*8 + {0..4}] = {mW, mB, vW, c2, vB}
// ---------------------------------------------------------------------------
__global__ __launch_bounds__(256) void feat_stats_kernel(
    const float* __restrict__ W,
    const float* __restrict__ Bb,
    float* __restrict__ st)
{
    const int wave = threadIdx.x >> 5;
    const int lane = threadIdx.x & 31;
    const int f = blockIdx.x * 8 + wave;
    if (f >= NUM_CONT) return;

    const v4f w = *(const v4f*)(W  + (size_t)f * EMB + lane * 4);
    const v4f b = *(const v4f*)(Bb + (size_t)f * EMB + lane * 4);

    float sw = (w.x + w.y) + (w.z + w.w);
    float sb = (b.x + b.y) + (b.z + b.w);
#pragma unroll
    for (int m = 16; m >= 1; m >>= 1) {
        sw += __shfl_xor(sw, m, 32);
        sb += __shfl_xor(sb, m, 32);
    }
    const float inv_d = 1.0f / (float)EMB;
    const float mW = sw * inv_d;
    const float mB = sb * inv_d;

    const float dw0 = w.x - mW, dw1 = w.y - mW, dw2 = w.z - mW, dw3 = w.w - mW;
    const float db0 = b.x - mB, db1 = b.y - mB, db2 = b.z - mB, db3 = b.w - mB;
    float vw = (dw0 * dw0 + dw1 * dw1) + (dw2 * dw2 + dw3 * dw3);
    float cb = (dw0 * db0 + dw1 * db1) + (dw2 * db2 + dw3 * db3);
    float vb = (db0 * db0 + db1 * db1) + (db2 * db2 + db3 * db3);
#pragma unroll
    for (int m = 16; m >= 1; m >>= 1) {
        vw += __shfl_xor(vw, m, 32);
        cb += __shfl_xor(cb, m, 32);
        vb += __shfl_xor(vb, m, 32);
    }
    if (lane == 0) {
        float* p = st + (size_t)f * 8;
        p[0] = mW;
        p[1] = mB;
        p[2] = vw * inv_d;
        p[3] = 2.0f * cb * inv_d;
        p[4] = vb * inv_d;
    }
}

// ---------------------------------------------------------------------------
// Kernel 2: main stream.  grid = (NUM_CONT, 8), block = 256 (8 wave32s).
// Each wave produces 16 rows of 128 floats with zero cross-lane reductions:
//   var  = vB + x*(c2 + x*vW)              (closed-form LayerNorm variance)
//   out  = (x*(W-mW) + (b-mB)) * rsqrt(var+eps) * gamma + beta
// ---------------------------------------------------------------------------
__global__ __launch_bounds__(256) void embed_ln_kernel(
    const float* __restrict__ x,      // [BATCH, NUM_CONT]
    const float* __restrict__ W,      // [NUM_CONT, EMB]
    const float* __restrict__ Bb,     // [NUM_CONT, EMB]
    const float* __restrict__ gamma,  // [EMB]
    const float* __restrict__ beta,   // [EMB]
    const float* __restrict__ st,     // [NUM_CONT, 8] moments
    float* __restrict__ out)          // [BATCH, NUM_CONT, EMB]
{
    __shared__ float smem[4 * EMB];   // W row | b row | gamma | beta (2 KB)

    const int f     = blockIdx.x;
    const int btile = blockIdx.y;           // 8 tiles x 128 batches
    const int wave  = threadIdx.x >> 5;     // 0..7
    const int lane  = threadIdx.x & 31;     // owns channels d = lane*4..+3

    // Wave-uniform per-feature moments (scalarizes to s_load).
    const float mW = st[(size_t)f * 8 + 0];
    const float mB = st[(size_t)f * 8 + 1];
    const float vW = st[(size_t)f * 8 + 2];
    const float c2 = st[(size_t)f * 8 + 3];
    const float vB = st[(size_t)f * 8 + 4];

    // ---- Stage parameters to LDS with the gfx1250 async-copy path ----------
    if (threadIdx.x < 32) {
        uint64_t gw = (uint64_t)(uintptr_t)(W     + (size_t)f * EMB + lane * 4);
        uint64_t gb = (uint64_t)(uintptr_t)(Bb    + (size_t)f * EMB + lane * 4);
        uint64_t gg = (uint64_t)(uintptr_t)(gamma + lane * 4);
        uint64_t ge = (uint64_t)(uintptr_t)(beta  + lane * 4);
        uint32_t lw = (uint32_t)(uintptr_t)(&smem[lane * 4]);
        uint32_t lb = lw + 1 * EMB * 4;
        uint32_t lg = lw + 2 * EMB * 4;
        uint32_t le = lw + 3 * EMB * 4;
        asm volatile(
            "global_load_async_to_lds_b128 %0, %4, off\n\t"
            "global_load_async_to_lds_b128 %1, %5, off\n\t"
            "global_load_async_to_lds_b128 %2, %6, off\n\t"
            "global_load_async_to_lds_b128 %3, %7, off\n\t"
            "s_wait_asynccnt 0"
            :
            : "v"(lw), "v"(lb), "v"(lg), "v"(le),
              "v"(gw), "v"(gb), "v"(gg), "v"(ge)
            : "memory");
    }
    __syncthreads();

    const v4f w4 = *(const v4f*)&smem[0 * EMB + lane * 4];
    const v4f b4 = *(const v4f*)&smem[1 * EMB + lane * 4];
    const v4f g4 = *(const v4f*)&smem[2 * EMB + lane * 4];
    const v4f t4 = *(const v4f*)&smem[3 * EMB + lane * 4];

    // Pre-centered loop-invariant registers: e - mean == x*wp + bp
    v4f wp, bp;
    wp.x = w4.x - mW; wp.y = w4.y - mW; wp.z = w4.z - mW; wp.w = w4.w - mW;
    bp.x = b4.x - mB; bp.y = b4.y - mB; bp.z = b4.z - mB; bp.w = b4.w - mB;

#pragma unroll 4
    for (int it = 0; it < 16; ++it) {
        const int batch = btile * 128 + wave * 16 + it;

        const float xv = x[(size_t)batch * NUM_CONT + f];

        // closed-form variance + rstd (3 VALU + rsqrt, no shuffles)
        const float var  = fmaf(xv, fmaf(xv, vW, c2), vB);
        const float rstd = rsqrtf(var + 1e-5f);

        // centered embedding
        const float u0 = fmaf(xv, wp.x, bp.x);
        const float u1 = fmaf(xv, wp.y, bp.y);
        const float u2 = fmaf(xv, wp.z, bp.z);
        const float u3 = fmaf(xv, wp.w, bp.w);

        v4f o;
        o.x = fmaf(u0 * rstd, g4.x, t4.x);
        o.y = fmaf(u1 * rstd, g4.y, t4.y);
        o.z = fmaf(u2 * rstd, g4.z, t4.z);
        o.w = fmaf(u3 * rstd, g4.w, t4.w);

        // 539 MB write-once stream: non-temporal 128-bit store.
        float* dst = out + ((size_t)batch * NUM_CONT + f) * EMB + lane * 4;
        __builtin_nontemporal_store(o, (v4f*)dst);
    }
}

extern "C" void kernel_launch(void* const* d_in, const int* in_sizes, int n_in,
                              void* d_out, int out_size, void* d_ws, size_t ws_size,
                              hipStream_t stream) {
    const float* x     = (const float*)d_in[0];
    const float* W     = (const float*)d_in[1];
    const float* b     = (const float*)d_in[2];
    const float* gamma = (const float*)d_in[3];
    const float* beta  = (const float*)d_in[4];
    float* out = (float*)d_out;
    float* st  = (float*)d_ws;   // NUM_CONT * 8 floats = 32 KB of scratch

    // Kernel 1: per-feature moments (1028 waves total).
    feat_stats_kernel<<<dim3((NUM_CONT + 7) / 8), 256, 0, stream>>>(W, b, st);

    // Kernel 2: main stream.
    dim3 grid(NUM_CONT, BATCH / 128);   // 1028 x 8 blocks
    embed_ln_kernel<<<grid, 256, 0, stream>>>(x, W, b, gamma, beta, st, out);
}